// DISK_35399120453934
// MI455X (gfx1250) — compile-verified
//
#include <hip/hip_runtime.h>
#include <hip/hip_bf16.h>
#include <math.h>

// ---------------------------------------------------------------------------
// SuperPoint-style keypoint extraction for (2,129,1024,1024) fp32 input.
// Outputs (flat in d_out, fp32): keypoints[2,2048,2], scores[2,2048],
// descriptors[2,2048,128].
// ---------------------------------------------------------------------------

#define W 1024
#define H 1024
#define HW (1024u * 1024u)
#define NB 2
#define KTOP 2048
#define DDIM 128
#define NCH 129

typedef float v2f __attribute__((ext_vector_type(2)));
typedef float v8f __attribute__((ext_vector_type(8)));

// ---------------- init: zero radix-select state ----------------
__global__ void kp_init(unsigned int* hist, unsigned long long* prefix,
                        unsigned int* kneed, unsigned int* cnt) {
  int t = threadIdx.x;
  hist[t] = 0u;
  hist[t + 256] = 0u;
  if (t < NB) {
    prefix[t] = 0ull;
    kneed[t] = KTOP;
    cnt[t] = 0u;
  }
}

// ---------------- NMS (5x5 SAME max-pool) + 64-bit key emit ----------------
__global__ void kp_nms_keys(const float* __restrict__ unet,
                            unsigned long long* __restrict__ keys) {
  int b = blockIdx.z;
  int bx = blockIdx.x * 32, by = blockIdx.y * 32;
  int tx = threadIdx.x, ty = threadIdx.y;              // 32 x 8 threads
  __shared__ float tile[36 * 36];
  const float* heat = unet + ((size_t)b * NCH + DDIM) * (size_t)HW;

  int tid = ty * 32 + tx;
  for (int i = tid; i < 36 * 36; i += 256) {
    int r = i / 36, c = i % 36;
    int gy = by + r - 2, gx = bx + c - 2;
    float v = -INFINITY;
    if (gy >= 0 && gy < H && gx >= 0 && gx < W) v = heat[gy * W + gx];
    tile[i] = v;
  }
  __syncthreads();

  for (int q = 0; q < 4; ++q) {
    int oy = ty + 8 * q;
    float hv = tile[(oy + 2) * 36 + tx + 2];
    float mx = -INFINITY;
#pragma unroll
    for (int dy = 0; dy < 5; ++dy)
#pragma unroll
      for (int dx = 0; dx < 5; ++dx)
        mx = fmaxf(mx, tile[(oy + dy) * 36 + tx + dx]);
    float s = (hv == mx) ? hv : 0.0f;                  // jnp.where(h==pooled,h,0)
    unsigned int u = __float_as_uint(s);
    unsigned int k32 = (u & 0x80000000u) ? ~u : (u | 0x80000000u); // monotonic
    unsigned int idx = (unsigned int)((by + oy) * W + bx + tx);
    // unique key: larger score first; ties -> smaller index first (JAX top_k)
    keys[((size_t)b << 20) + idx] =
        ((unsigned long long)k32 << 32) | (unsigned long long)(~idx);
  }
}

// ---------------- radix-select: histogram of current digit ----------------
__global__ void kp_hist(const unsigned long long* __restrict__ keys,
                        const unsigned long long* __restrict__ prefix,
                        unsigned int* __restrict__ hist, int shift) {
  __shared__ unsigned int lh[512];
  int t = threadIdx.x;
  lh[t] = 0u;
  lh[t + 256] = 0u;
  __syncthreads();
  unsigned int i = blockIdx.x * 256 + t;               // 8192 blocks -> 2*2^20
  int b = i >> 20;
  unsigned long long key = keys[i];
  unsigned long long pf = prefix[b];
  if ((((key ^ pf) >> shift) >> 8) == 0ull) {          // high bits match prefix
    unsigned int d = (unsigned int)((key >> shift) & 255ull);
    atomicAdd(&lh[(b << 8) + d], 1u);
  }
  __syncthreads();
  if (lh[t]) atomicAdd(&hist[t], lh[t]);
  if (lh[t + 256]) atomicAdd(&hist[t + 256], lh[t + 256]);
}

// ---------------- radix-select: pick digit, update prefix/kneed ----------------
__global__ void kp_select(unsigned int* __restrict__ hist,
                          unsigned long long* __restrict__ prefix,
                          unsigned int* __restrict__ kneed, int shift) {
  int b = blockIdx.x;
  int t = threadIdx.x;
  __shared__ unsigned int lh[256];
  lh[t] = hist[(b << 8) + t];
  __syncthreads();
  hist[(b << 8) + t] = 0u;                             // ready for next pass
  if (t == 0) {
    unsigned int need = kneed[b];
    unsigned int total = 0;
    int chosen = 0;
    for (int d = 255; d >= 0; --d) {
      unsigned int c = lh[d];
      if (total + c >= need) { chosen = d; break; }
      total += c;
    }
    kneed[b] = need - total;
    prefix[b] |= ((unsigned long long)chosen) << shift;
  }
}

// ---------------- compact: gather all keys >= threshold (exactly KTOP) ------
__global__ void kp_compact(const unsigned long long* __restrict__ keys,
                           const unsigned long long* __restrict__ prefix,
                           unsigned long long* __restrict__ sel,
                           unsigned int* __restrict__ cnt) {
  unsigned int i = blockIdx.x * 256 + threadIdx.x;
  int b = i >> 20;
  unsigned long long key = keys[i];
  if (key >= prefix[b]) {                              // prefix == exact k-th key
    unsigned int pos = atomicAdd(&cnt[b], 1u);
    if (pos < KTOP) sel[((unsigned int)b << 11) + pos] = key;
  }
}

// ---------------- bitonic sort (descending) + decode outputs ----------------
__global__ void kp_sort_out(const unsigned long long* __restrict__ sel,
                            unsigned int* __restrict__ sel_idx,
                            float* __restrict__ kp, float* __restrict__ scores) {
  int b = blockIdx.x;
  int t = threadIdx.x;                                 // 1024 threads
  __shared__ unsigned long long s[KTOP];
  s[t] = sel[(b << 11) + t];
  s[t + 1024] = sel[(b << 11) + t + 1024];
  for (int k = 2; k <= KTOP; k <<= 1) {
    for (int j = k >> 1; j > 0; j >>= 1) {
      __syncthreads();
      for (int e = t; e < KTOP; e += 1024) {
        int p = e ^ j;
        if (p > e) {
          unsigned long long a = s[e], c = s[p];
          bool desc_blk = ((e & k) == 0);
          bool doswap = desc_blk ? (a < c) : (a > c);  // overall descending
          if (doswap) { s[e] = c; s[p] = a; }
        }
      }
    }
  }
  __syncthreads();
  for (int e = t; e < KTOP; e += 1024) {
    unsigned long long key = s[e];
    unsigned int idx = ~((unsigned int)key);
    unsigned int k32 = (unsigned int)(key >> 32);
    unsigned int u = (k32 & 0x80000000u) ? (k32 & 0x7FFFFFFFu) : ~k32;
    float sc = __uint_as_float(u);
    int x = (int)(idx & (W - 1));
    int y = (int)((idx >> 10) & (H - 1));
    size_t r = ((size_t)b << 11) + e;
    kp[r * 2 + 0] = (float)x;
    kp[r * 2 + 1] = (float)y;
    scores[r] = sc;
    sel_idx[r] = idx;
  }
}

// ---------------- descriptor gather + WMMA Gram-diag L2 normalize -----------
// One wave (32 lanes) handles 16 keypoints. X = 16x128 descriptor block.
// Squared row norms = diag(X * X^T), accumulated via 32 chained
// V_WMMA_F32_16X16X4_F32 ops. Per the CDNA5 f32 A(16x4)/B(4x16) layouts the
// per-lane registers of X-as-A and X^T-as-B coincide, so B operand == A.
__global__ void kp_desc_norm(const float* __restrict__ unet,
                             const unsigned int* __restrict__ sel_idx,
                             float* __restrict__ out_desc) {
  int b = blockIdx.y;
  int base = blockIdx.x << 4;                          // 16 keypoints per block
  int l = threadIdx.x;                                 // 0..31
  int m = l & 15;                                      // row (keypoint in group)
  int hh = l >> 4;                                     // K-half selector
  unsigned int idx = sel_idx[(b << 11) + base + m];
  const float* db = unet + (size_t)b * NCH * (size_t)HW + idx;

  v2f xv[32];
  v8f acc = {0.f, 0.f, 0.f, 0.f, 0.f, 0.f, 0.f, 0.f};
#pragma unroll
  for (int k = 0; k < 32; ++k) {
    int j = 4 * k + 2 * hh;                            // channel pair for this lane
    v2f a;
    a.x = db[(size_t)j * HW];
    a.y = db[(size_t)(j + 1) * HW];
    xv[k] = a;
    acc = __builtin_amdgcn_wmma_f32_16x16x4_f32(
        /*neg_a=*/false, a, /*neg_b=*/false, a,
        /*c_mod=*/(short)0, acc, /*reuse_a=*/false, /*reuse_b=*/false);
  }

  __shared__ float gram[256];
#pragma unroll
  for (int r = 0; r < 8; ++r) {
    int M = r + (hh << 3);                             // C/D layout: M=r(+8), N=lane%16
    gram[M * 16 + m] = acc[r];
  }
  __syncthreads();

  float inv = 1.0f / fmaxf(sqrtf(gram[m * 16 + m]), 1e-12f);
  float* ob = out_desc + ((size_t)((b << 11) + base + m)) * DDIM;
#pragma unroll
  for (int k = 0; k < 32; ++k) {
    int j = 4 * k + 2 * hh;
    ob[j]     = xv[k].x * inv;
    ob[j + 1] = xv[k].y * inv;
  }
}

// ---------------------------------------------------------------------------
extern "C" void kernel_launch(void* const* d_in, const int* in_sizes, int n_in,
                              void* d_out, int out_size, void* d_ws, size_t ws_size,
                              hipStream_t stream) {
  const float* unet = (const float*)d_in[0];

  // workspace layout (bytes)
  char* ws = (char*)d_ws;
  unsigned long long* keys   = (unsigned long long*)(ws);                // 16 MB
  unsigned long long* sel    = (unsigned long long*)(ws + 16777216);     // 32 KB
  unsigned int*       selidx = (unsigned int*)(ws + 16777216 + 32768);   // 16 KB
  unsigned int*       hist   = (unsigned int*)(ws + 16777216 + 49152);   // 2 KB
  unsigned long long* prefix = (unsigned long long*)(ws + 16777216 + 51200);
  unsigned int*       kneed  = (unsigned int*)(ws + 16777216 + 51216);
  unsigned int*       cnt    = (unsigned int*)(ws + 16777216 + 51224);

  float* kp_out    = (float*)d_out;                    // [2,2048,2]
  float* score_out = kp_out + (size_t)NB * KTOP * 2;   // [2,2048]
  float* desc_out  = score_out + (size_t)NB * KTOP;    // [2,2048,128]

  kp_init<<<1, 256, 0, stream>>>(hist, prefix, kneed, cnt);

  kp_nms_keys<<<dim3(W / 32, H / 32, NB), dim3(32, 8, 1), 0, stream>>>(unet, keys);

  for (int pass = 0; pass < 8; ++pass) {
    int shift = 56 - 8 * pass;
    kp_hist<<<8192, 256, 0, stream>>>(keys, prefix, hist, shift);
    kp_select<<<NB, 256, 0, stream>>>(hist, prefix, kneed, shift);
  }

  kp_compact<<<8192, 256, 0, stream>>>(keys, prefix, sel, cnt);

  kp_sort_out<<<NB, 1024, 0, stream>>>(sel, selidx, kp_out, score_out);

  kp_desc_norm<<<dim3(KTOP / 16, NB, 1), 32, 0, stream>>>(unet, selidx, desc_out);
}